// MARGINLossHead_61615600828700
// MI455X (gfx1250) — compile-verified
//
#include <hip/hip_runtime.h>
#include <stdint.h>

#ifndef __has_builtin
#define __has_builtin(x) 0
#endif

namespace {

constexpr int   kB       = 2048;
constexpr int   kC       = 50000;
constexpr int   kV4      = kC / 4;                 // 12500 float4 per row (exact)
constexpr int   kT       = 256;                    // threads per block (8 waves)
constexpr int   kUnroll  = 4;                      // float4 async loads / thread / stage
constexpr int   kStageV4 = kT * kUnroll;           // 1024 float4 = 16 KB per stage
constexpr int   kStages  = (kV4 + kStageV4 - 1) / kStageV4;  // 13
constexpr int   kNBuf    = 4;                      // LDS ring buffers (64 KB total)
constexpr int   kPre     = 3;                      // pipeline depth (< kNBuf)

constexpr float kEps   = 1e-7f;
constexpr float kScale = 64.0f;
constexpr float kK     = 92.332482616893656f;      // 64 * log2(e)
constexpr float kLn2   = 0.69314718055994531f;

// b128 async builtin wants int4-typed pointers (per hipcc diagnostic):
//   param0: int __vector(4) addrspace(1)*  (global source)
//   param1: int __vector(4) addrspace(3)*  (LDS destination)
typedef int v4i __attribute__((vector_size(16)));
typedef __attribute__((address_space(1))) v4i glb_v4i_t;
typedef __attribute__((address_space(3))) v4i lds_v4i_t;
typedef __attribute__((address_space(3))) char lds_char_t;

__device__ __forceinline__ void async_copy_b128(const void* gsrc, void* ldst) {
#if __has_builtin(__builtin_amdgcn_global_load_async_to_lds_b128)
  __builtin_amdgcn_global_load_async_to_lds_b128(
      (glb_v4i_t*)gsrc, (lds_v4i_t*)ldst, 0, 0);
#else
  unsigned lds_off = (unsigned)(uintptr_t)(lds_char_t*)ldst;
  unsigned long long gaddr = (unsigned long long)(uintptr_t)gsrc;
  asm volatile("global_load_async_to_lds_b128 %0, %1, off"
               :: "v"(lds_off), "v"(gaddr)
               : "memory");
#endif
}

template <int N>
__device__ __forceinline__ void wait_asynccnt() {
#if __has_builtin(__builtin_amdgcn_s_wait_asynccnt)
  __builtin_amdgcn_s_wait_asynccnt(N);
  asm volatile("" ::: "memory");
#else
  asm volatile("s_wait_asynccnt %0" :: "i"(N) : "memory");
#endif
}

__device__ __forceinline__ void wait_dscnt0() {
#if __has_builtin(__builtin_amdgcn_s_wait_dscnt)
  __builtin_amdgcn_s_wait_dscnt(0);
  asm volatile("" ::: "memory");
#else
  asm volatile("s_wait_dscnt 0x0" ::: "memory");
#endif
}

__device__ __forceinline__ float fast_exp2(float x) {
#if __has_builtin(__builtin_amdgcn_exp2f)
  return __builtin_amdgcn_exp2f(x);
#else
  return exp2f(x);
#endif
}

__device__ __forceinline__ float fast_log2(float x) {
#if __has_builtin(__builtin_amdgcn_logf)
  return __builtin_amdgcn_logf(x);
#else
  return log2f(x);
#endif
}

__device__ __forceinline__ float clip_cos(float x) {
  return fminf(fmaxf(x, -1.0f + kEps), 1.0f - kEps);
}

// exp(64*(v-1)) = exp2(v*64*log2e - 64*log2e): fixed-shift softmax term.
// Valid because clipped cosines bound logits to (-64, 64).
__device__ __forceinline__ float shifted_exp(float v) {
  return fast_exp2(fmaf(v, kK, -kK));
}

__global__ __launch_bounds__(kT) void arcface_row_lse(
    const float* __restrict__ cos_theta,
    const int*   __restrict__ labels,
    const float* __restrict__ margins,
    float*       __restrict__ row_loss)
{
  // 64 KB LDS ring of thread-private FIFO slots (no cross-thread sharing
  // during streaming -> no barriers in the hot loop).
  __shared__ float4 stage[kNBuf][kStageV4];

  const int b = blockIdx.x;
  const int t = threadIdx.x;
  const float* __restrict__ row = cos_theta + (size_t)b * kC;

  // Label-column fixup data (thread 0 only; latency hidden under the stream).
  float ct_label = 0.0f, target = 0.0f;
  if (t == 0) {
    const int   lab = labels[b];
    const float m   = margins[lab];
    const float cm  = cosf(m);
    const float sm  = sinf(m);
    ct_label = clip_cos(row[lab]);
    const float st = sqrtf(fmaxf(1.0f - ct_label * ct_label, 0.0f));
    target = ct_label * cm - st * sm;              // cos(theta + m)
  }

  auto issue = [&](int s) {
    const int slot = s & (kNBuf - 1);
#pragma unroll
    for (int u = 0; u < kUnroll; ++u) {
      int idx = s * kStageV4 + u * kT + t;
      idx = idx < kV4 ? idx : (kV4 - 1);           // clamp keeps ASYNCcnt exact
      async_copy_b128(((const float4*)row) + idx, &stage[slot][u * kT + t]);
    }
  };

  for (int s = 0; s < kPre; ++s) issue(s);         // kStages(13) > kPre(3)

  float acc = 0.0f;
#pragma unroll 1
  for (int s = 0; s < kStages; ++s) {
    // Oldest of K in-flight stages is done when ASYNCcnt <= 4*(K-1)
    // (async loads complete in order; 4 per-wave issues per stage).
    const int inflight = (kStages - s) < kPre ? (kStages - s) : kPre;
    if (inflight >= 3)      wait_asynccnt<2 * kUnroll>();
    else if (inflight == 2) wait_asynccnt<1 * kUnroll>();
    else                    wait_asynccnt<0>();

    const int slot = s & (kNBuf - 1);
#pragma unroll
    for (int u = 0; u < kUnroll; ++u) {
      const int    idx = s * kStageV4 + u * kT + t;
      const float4 v   = stage[slot][u * kT + t];  // ds_load_b128
      if (idx < kV4) {
        acc += shifted_exp(clip_cos(v.x));
        acc += shifted_exp(clip_cos(v.y));
        acc += shifted_exp(clip_cos(v.z));
        acc += shifted_exp(clip_cos(v.w));
      }
    }
    wait_dscnt0();                 // LDS reads retired before DMA reuses a slot
    if (s + kPre < kStages) issue(s + kPre);
  }

  // Block reduction (alias stage memory; all async/ds traffic drained above).
  __syncthreads();
  float* red = (float*)&stage[0][0];
  red[t] = acc;
  __syncthreads();
  for (int off = kT / 2; off > 0; off >>= 1) {
    if (t < off) red[t] += red[t + off];
    __syncthreads();
  }

  if (t == 0) {
    float S = red[0];
    // Swap plain label term for the margin term.
    S += shifted_exp(target) - shifted_exp(ct_label);
    // loss_b = logsumexp - target_logit ; logsumexp = 64 + ln(S)
    row_loss[b] = kScale + fast_log2(S) * kLn2 - kScale * target;
  }
}

__global__ __launch_bounds__(256) void reduce_mean(
    const float* __restrict__ row_loss, float* __restrict__ out)
{
  __shared__ float red[256];
  float a = 0.0f;
  for (int i = threadIdx.x; i < kB; i += 256) a += row_loss[i];
  red[threadIdx.x] = a;
  __syncthreads();
  for (int off = 128; off > 0; off >>= 1) {
    if (threadIdx.x < off) red[threadIdx.x] += red[threadIdx.x + off];
    __syncthreads();
  }
  if (threadIdx.x == 0) out[0] = red[0] * (1.0f / (float)kB);
}

}  // namespace

extern "C" void kernel_launch(void* const* d_in, const int* in_sizes, int n_in,
                              void* d_out, int out_size, void* d_ws, size_t ws_size,
                              hipStream_t stream) {
  (void)in_sizes; (void)n_in; (void)out_size; (void)ws_size;
  const float* cos_theta = (const float*)d_in[0];
  const int*   labels    = (const int*)d_in[1];
  const float* margins   = (const float*)d_in[2];
  float* out      = (float*)d_out;
  float* row_loss = (float*)d_ws;   // 2048 floats of scratch

  arcface_row_lse<<<kB, kT, 0, stream>>>(cos_theta, labels, margins, row_loss);
  reduce_mean<<<1, 256, 0, stream>>>(row_loss, out);
}